// RepresentationNetwork_52338471469712
// MI455X (gfx1250) — compile-verified
//
#include <hip/hip_runtime.h>
#include <hip/hip_bf16.h>
#include <math.h>

#define B_    64
#define L_    1024
#define D_    16
#define H_    128
#define TOPK_ 64
#define SLABS 32
#define ROWS  32                     // query rows per slab (per WG in pass 1)
#define SCALE_ 0.08838834764831845f  // 128^-0.5

typedef float v2f __attribute__((ext_vector_type(2)));
typedef float v8f __attribute__((ext_vector_type(8)));

// ---------------------------------------------------------------------------
// Kernel 0: Q = x@Wq + bq, K = x@Wk + bk, mask[b,l] = (sum|x| != 0)
// grid: B*L blocks of H threads
// ---------------------------------------------------------------------------
__global__ __launch_bounds__(H_) void qk_proj_kernel(
    const float* __restrict__ x,
    const float* __restrict__ Wq, const float* __restrict__ bq,
    const float* __restrict__ Wk, const float* __restrict__ bk,
    float* __restrict__ Q, float* __restrict__ K, unsigned* __restrict__ mask) {
  const int row = blockIdx.x;   // b*L + l
  const int h = threadIdx.x;    // 0..127
  __shared__ float xr[D_];
  if (h < D_) xr[h] = x[(size_t)row * D_ + h];
  __syncthreads();
  float q = bq[h], k = bk[h];
#pragma unroll
  for (int d = 0; d < D_; ++d) {
    q = fmaf(xr[d], Wq[d * H_ + h], q);
    k = fmaf(xr[d], Wk[d * H_ + h], k);
  }
  Q[(size_t)row * H_ + h] = q;
  K[(size_t)row * H_ + h] = k;
  if (h == 0) {
    float s = 0.f;
#pragma unroll
    for (int d = 0; d < D_; ++d) s += fabsf(xr[d]);
    mask[row] = (s != 0.f) ? 1u : 0u;
  }
}

// ---------------------------------------------------------------------------
// Kernel 1: scores slab (ROWS x L) via WMMA f32 16x16x4, then local top-64.
// grid: (SLABS, B), 256 threads (8 waves). Dynamic LDS: ROWS*L_*4 = 128KB.
// Wave w owns col-tiles ct = w*8 .. w*8+7 and BOTH row tiles (rt0, rt1):
//   two A caches (128 VGPRs/lane) -> each B fragment load feeds 2 WMMAs.
// A frag (16x4 f32): lane<16 -> M=lane, k={k0,k0+1}; lane>=16 -> M=lane-16, k={k0+2,k0+3}
// B frag (4x16 f32): lane<16 -> N=lane, k={k0,k0+1}; lane>=16 -> N=lane-16, k={k0+2,k0+3}
// C/D (16x16 f32):   elem e: lane<16 -> (M=e,   N=lane); lane>=16 -> (M=e+8, N=lane-16)
// ---------------------------------------------------------------------------
__global__ __launch_bounds__(256) void scores_topk_kernel(
    const float* __restrict__ Q, const float* __restrict__ K,
    const unsigned* __restrict__ mask,
    float* __restrict__ cand_val, unsigned* __restrict__ cand_idx) {
  extern __shared__ float ssc[];  // ROWS * L_ scores
  __shared__ float sval[256];     // per-thread cached chunk max
  __shared__ unsigned spos[256];
  __shared__ float wval[256];     // tree-reduce workspace
  __shared__ unsigned wpos[256];

  const int slab = blockIdx.x;
  const int b = blockIdx.y;
  const int tid = threadIdx.x;
  const int wave = tid >> 5;
  const int lane = tid & 31;
  const int m15 = lane & 15;
  const int koff = (lane >> 4) * 2;

  // --- GEMM phase ---
  const float* qbase0 =
      Q + ((size_t)(b * L_ + slab * ROWS + m15)) * H_ + koff;       // rt=0
  const float* qbase1 = qbase0 + 16 * H_;                           // rt=1
  v2f a0[32], a1[32];
#pragma unroll
  for (int ks = 0; ks < 32; ++ks) {
    a0[ks] = *(const v2f*)(qbase0 + ks * 4);
    a1[ks] = *(const v2f*)(qbase1 + ks * 4);
  }

  const int ctBase = wave * 8;
  for (int i = 0; i < 8; ++i) {
    const int ct = ctBase + i;
    const float* kbase = K + ((size_t)(b * L_ + ct * 16 + m15)) * H_ + koff;
    v8f c0 = {0.f, 0.f, 0.f, 0.f, 0.f, 0.f, 0.f, 0.f};
    v8f c1 = {0.f, 0.f, 0.f, 0.f, 0.f, 0.f, 0.f, 0.f};
#pragma unroll
    for (int ks = 0; ks < 32; ++ks) {
      v2f bf = *(const v2f*)(kbase + ks * 4);
      c0 = __builtin_amdgcn_wmma_f32_16x16x4_f32(false, a0[ks], false, bf,
                                                 (short)0, c0, false, false);
      c1 = __builtin_amdgcn_wmma_f32_16x16x4_f32(false, a1[ks], false, bf,
                                                 (short)0, c1, false, false);
    }
    // scale + mask + stage to LDS
    const int n = lane & 15;
    const int mb = (lane >> 4) * 8;
    const int col = ct * 16 + n;
    const unsigned mc = mask[b * L_ + col];
#pragma unroll
    for (int e = 0; e < 8; ++e) {
      const int rl0 = mb + e;        // row within slab, tile 0
      const int rl1 = 16 + mb + e;   // row within slab, tile 1
      const unsigned mr0 = mask[b * L_ + slab * ROWS + rl0];
      const unsigned mr1 = mask[b * L_ + slab * ROWS + rl1];
      float v0 = c0[e] * SCALE_;
      float v1 = c1[e] * SCALE_;
      if (!(mr0 & mc)) v0 = -INFINITY;
      if (!(mr1 & mc)) v1 = -INFINITY;
      ssc[rl0 * L_ + col] = v0;
      ssc[rl1 * L_ + col] = v1;
    }
  }
  __syncthreads();

  // --- selection phase: cached chunk maxima + tree reduce per round ---
  const int CH = (ROWS * L_) / 256;  // 128 elements per thread chunk
  {
    const unsigned base = (unsigned)tid * CH;
    float best = ssc[base];
    unsigned bp = base;
    for (int t = 1; t < CH; ++t) {
      const unsigned p = base + t;
      const float v = ssc[p];
      if (v > best) { best = v; bp = p; }  // strict > keeps lowest index on tie
    }
    sval[tid] = best;
    spos[tid] = bp;
  }
  __syncthreads();

  for (int iter = 0; iter < TOPK_; ++iter) {
    wval[tid] = sval[tid];
    wpos[tid] = spos[tid];
    __syncthreads();
#pragma unroll
    for (int off = 128; off > 0; off >>= 1) {
      if (tid < off) {
        const float v2 = wval[tid + off];
        const unsigned p2 = wpos[tid + off];
        if (v2 > wval[tid] || (v2 == wval[tid] && p2 < wpos[tid])) {
          wval[tid] = v2;
          wpos[tid] = p2;
        }
      }
      __syncthreads();
    }
    const float bv = wval[0];
    const unsigned bp = wpos[0];
    if (tid == 0) {
      const size_t o = ((size_t)(b * SLABS + slab)) * TOPK_ + iter;
      cand_val[o] = bv;
      cand_idx[o] = (unsigned)(slab * ROWS) * L_ + bp;  // flat index in L*L
    }
    if (tid == (int)(bp >> 7)) {  // owner of winner chunk: remove + rescan
      ssc[bp] = -INFINITY;
      const unsigned base = (unsigned)tid * CH;
      float best = ssc[base];
      unsigned q = base;
      for (int t = 1; t < CH; ++t) {
        const unsigned p = base + t;
        const float v = ssc[p];
        if (v > best) { best = v; q = p; }
      }
      sval[tid] = best;
      spos[tid] = q;
    }
    __syncthreads();
  }
}

// ---------------------------------------------------------------------------
// Kernel 2: merge 32*64 candidates -> exact top-64 per batch, softmax weights.
// grid: B blocks of 256 threads. Tie-break by global flat index (lax.top_k).
// ---------------------------------------------------------------------------
__global__ __launch_bounds__(256) void merge_softmax_kernel(
    const float* __restrict__ cand_val, const unsigned* __restrict__ cand_idx,
    int* __restrict__ sel_row, int* __restrict__ sel_col,
    float* __restrict__ sel_w) {
  const int b = blockIdx.x;
  const int tid = threadIdx.x;
  const int NC = SLABS * TOPK_;  // 2048
  const int CH = NC / 256;       // 8
  __shared__ float v[SLABS * TOPK_];
  __shared__ unsigned id[SLABS * TOPK_];
  __shared__ float sval[256];
  __shared__ unsigned spos[256];
  __shared__ unsigned sflat[256];
  __shared__ float wval[256];
  __shared__ unsigned wpos[256];
  __shared__ unsigned wflat[256];
  __shared__ float selv[TOPK_];
  __shared__ unsigned selid[TOPK_];
  __shared__ float es[TOPK_ + 1];

  for (int t = tid; t < NC; t += 256) {
    v[t] = cand_val[(size_t)b * NC + t];
    id[t] = cand_idx[(size_t)b * NC + t];
  }
  __syncthreads();

  {
    const int base = tid * CH;
    float best = v[base];
    unsigned bp = (unsigned)base, bf = id[base];
#pragma unroll
    for (int t = 1; t < CH; ++t) {
      const int p = base + t;
      const float val = v[p];
      const unsigned f = id[p];
      if (val > best || (val == best && f < bf)) {
        best = val; bp = (unsigned)p; bf = f;
      }
    }
    sval[tid] = best;
    spos[tid] = bp;
    sflat[tid] = bf;
  }
  __syncthreads();

  for (int iter = 0; iter < TOPK_; ++iter) {
    wval[tid] = sval[tid];
    wpos[tid] = spos[tid];
    wflat[tid] = sflat[tid];
    __syncthreads();
#pragma unroll
    for (int off = 128; off > 0; off >>= 1) {
      if (tid < off) {
        const float v2 = wval[tid + off];
        if (v2 > wval[tid] ||
            (v2 == wval[tid] && wflat[tid + off] < wflat[tid])) {
          wval[tid] = v2;
          wpos[tid] = wpos[tid + off];
          wflat[tid] = wflat[tid + off];
        }
      }
      __syncthreads();
    }
    const unsigned bp = wpos[0];
    if (tid == 0) {
      selv[iter] = wval[0];
      selid[iter] = wflat[0];
    }
    if (tid == (int)(bp >> 3)) {  // owner chunk (CH=8): remove + rescan
      v[bp] = -INFINITY;
      const int base = tid * CH;
      float best = v[base];
      unsigned q = (unsigned)base, bf = id[base];
#pragma unroll
      for (int t = 1; t < CH; ++t) {
        const int p = base + t;
        const float val = v[p];
        const unsigned f = id[p];
        if (val > best || (val == best && f < bf)) {
          best = val; q = (unsigned)p; bf = f;
        }
      }
      sval[tid] = best;
      spos[tid] = q;
      sflat[tid] = bf;
    }
    __syncthreads();
  }

  // stable softmax (selv[0] is the max)
  if (tid < TOPK_) es[tid] = expf(selv[tid] - selv[0]);
  __syncthreads();
  if (tid == 0) {
    float s = 0.f;
    for (int k = 0; k < TOPK_; ++k) s += es[k];
    es[TOPK_] = s;
  }
  __syncthreads();
  if (tid < TOPK_) {
    const unsigned f = selid[tid];
    sel_row[b * TOPK_ + tid] = (int)(f / L_);
    sel_col[b * TOPK_ + tid] = (int)(f % L_);
    sel_w[b * TOPK_ + tid] = es[tid] / es[TOPK_];
  }
}

// ---------------------------------------------------------------------------
// Kernel 3: gather pairs, phi/xi MLPs, weighted pool, rho head.
// grid: B blocks of H threads (thread = output channel)
// ---------------------------------------------------------------------------
__global__ __launch_bounds__(H_) void interact_pool_kernel(
    const float* __restrict__ x,
    const float* __restrict__ phi_W1, const float* __restrict__ phi_b1,
    const float* __restrict__ phi_W2, const float* __restrict__ phi_b2,
    const float* __restrict__ xi_W1, const float* __restrict__ xi_b1,
    const float* __restrict__ xi_W2, const float* __restrict__ xi_b2,
    const float* __restrict__ rho_W1, const float* __restrict__ rho_b1,
    const float* __restrict__ rho_W2, const float* __restrict__ rho_b2,
    const int* __restrict__ sel_row, const int* __restrict__ sel_col,
    const float* __restrict__ sel_w, float* __restrict__ out) {
  const int b = blockIdx.x;
  const int h = threadIdx.x;
  __shared__ float cat[2 * D_];
  __shared__ float hid[H_];
  __shared__ float pooledS[H_];

  float pooled = 0.f;
  for (int k = 0; k < TOPK_; ++k) {
    const int r = sel_row[b * TOPK_ + k];
    const int c = sel_col[b * TOPK_ + k];
    const float w = sel_w[b * TOPK_ + k];
    if (h < D_) cat[h] = x[((size_t)b * L_ + r) * D_ + h];
    else if (h < 2 * D_) cat[h] = x[((size_t)b * L_ + c) * D_ + (h - D_)];
    __syncthreads();
    float o;
    if (r == c) {  // uniform branch across the block
      float a = phi_b1[h];
#pragma unroll
      for (int d = 0; d < D_; ++d) a = fmaf(cat[d], phi_W1[d * H_ + h], a);
      hid[h] = fmaxf(a, 0.f);
      __syncthreads();
      o = phi_b2[h];
      for (int j = 0; j < H_; ++j) o = fmaf(hid[j], phi_W2[j * H_ + h], o);
    } else {
      float a = xi_b1[h];
#pragma unroll
      for (int d = 0; d < 2 * D_; ++d) a = fmaf(cat[d], xi_W1[d * H_ + h], a);
      hid[h] = fmaxf(a, 0.f);
      __syncthreads();
      o = xi_b2[h];
      for (int j = 0; j < H_; ++j) o = fmaf(hid[j], xi_W2[j * H_ + h], o);
    }
    pooled = fmaf(w, o, pooled);
    __syncthreads();  // before cat/hid are overwritten next iteration
  }

  pooledS[h] = pooled;
  __syncthreads();
  float a = rho_b1[h];
  for (int j = 0; j < H_; ++j) a = fmaf(pooledS[j], rho_W1[j * H_ + h], a);
  hid[h] = fmaxf(a, 0.f);
  __syncthreads();
  float o = rho_b2[h];
  for (int j = 0; j < H_; ++j) o = fmaf(hid[j], rho_W2[j * H_ + h], o);
  out[b * H_ + h] = o;
}

// ---------------------------------------------------------------------------
extern "C" void kernel_launch(void* const* d_in, const int* in_sizes, int n_in,
                              void* d_out, int out_size, void* d_ws,
                              size_t ws_size, hipStream_t stream) {
  const float* x      = (const float*)d_in[0];
  const float* Wq     = (const float*)d_in[1];
  const float* bq     = (const float*)d_in[2];
  const float* Wk     = (const float*)d_in[3];
  const float* bk     = (const float*)d_in[4];
  const float* phi_W1 = (const float*)d_in[5];
  const float* phi_b1 = (const float*)d_in[6];
  const float* phi_W2 = (const float*)d_in[7];
  const float* phi_b2 = (const float*)d_in[8];
  const float* xi_W1  = (const float*)d_in[9];
  const float* xi_b1  = (const float*)d_in[10];
  const float* xi_W2  = (const float*)d_in[11];
  const float* xi_b2  = (const float*)d_in[12];
  const float* rho_W1 = (const float*)d_in[13];
  const float* rho_b1 = (const float*)d_in[14];
  const float* rho_W2 = (const float*)d_in[15];
  const float* rho_b2 = (const float*)d_in[16];
  float* outF = (float*)d_out;

  char* ws = (char*)d_ws;
  const size_t szQK = (size_t)B_ * L_ * H_ * sizeof(float);     // 32 MB each
  float* Q = (float*)(ws);
  float* K = (float*)(ws + szQK);
  unsigned* mask = (unsigned*)(ws + 2 * szQK);
  const size_t offCV = 2 * szQK + (size_t)B_ * L_ * sizeof(unsigned);
  const size_t szC = (size_t)B_ * SLABS * TOPK_ * sizeof(float);
  float* cand_val = (float*)(ws + offCV);
  unsigned* cand_idx = (unsigned*)(ws + offCV + szC);
  const size_t offSel = offCV + 2 * szC;
  int* sel_row = (int*)(ws + offSel);
  int* sel_col = (int*)(ws + offSel + (size_t)B_ * TOPK_ * 4);
  float* sel_w = (float*)(ws + offSel + 2 * (size_t)B_ * TOPK_ * 4);

  qk_proj_kernel<<<dim3(B_ * L_), dim3(H_), 0, stream>>>(x, Wq, bq, Wk, bk, Q,
                                                         K, mask);
  const size_t smemScores = (size_t)ROWS * L_ * sizeof(float);  // 128 KB
  scores_topk_kernel<<<dim3(SLABS, B_), dim3(256), smemScores, stream>>>(
      Q, K, mask, cand_val, cand_idx);
  merge_softmax_kernel<<<dim3(B_), dim3(256), 0, stream>>>(
      cand_val, cand_idx, sel_row, sel_col, sel_w);
  interact_pool_kernel<<<dim3(B_), dim3(H_), 0, stream>>>(
      x, phi_W1, phi_b1, phi_W2, phi_b2, xi_W1, xi_b1, xi_W2, xi_b2, rho_W1,
      rho_b1, rho_W2, rho_b2, sel_row, sel_col, sel_w, outF);
}